// Matcher_75419625718226
// MI455X (gfx1250) — compile-verified
//
#include <hip/hip_runtime.h>
#include <stdint.h>

// Batched square Jonker-Volgenant LAP for MI455X (gfx1250, wave32).
//
// Design (reasoned from CDNA5 specs, compile-only):
//  * One batch per workgroup; the full 256x256 f32 cost matrix (256KB) lives
//    in LDS (CDNA5: up to 320KB per workgroup, ISA 3.1/4.3.4) with row stride
//    257 dwords -> 257 % 64 == 1 gives conflict-free transposed column reads.
//  * Staging uses the Tensor Data Mover (tensor_load_to_lds) with
//    pad_interval=256dw / pad_amount=1dw so the DMA itself inserts the
//    padding; completion via s_wait_tensorcnt 0 in the issuing wave.
//  * The 270KB LDS footprint pins one workgroup per WGP, so extra waves buy
//    no occupancy -- instead ONE wave32 runs the whole solver wave-
//    synchronously: zero barriers on the serial Dijkstra critical path.
//    Lane L owns columns/rows {L+32k}; argmin over 256 = 8 register compares
//    + 5-step __shfl_xor butterfly (lowest-index tie-break == jnp.argmin).
//  * WMMA is not applicable: JV-LAP contains no dense matmul.

#define INF_F 1e30f

constexpr int NN = 256;       // problem size (reference fixes N=256)
constexpr int NS = 257;       // padded LDS row stride in dwords
constexpr int K  = 8;         // columns/rows owned per lane (256/32)

typedef unsigned int u32x4 __attribute__((ext_vector_type(4)));
typedef int          i32x4 __attribute__((ext_vector_type(4)));
typedef int          i32x8 __attribute__((ext_vector_type(8)));

__global__ __launch_bounds__(32, 1)
void Matcher_75419625718226_kernel(const float* __restrict__ costs,
                                   const void* __restrict__ mask_raw,
                                   int* __restrict__ out)
{
    extern __shared__ unsigned char smem[];
    float* C        = (float*)smem;             // [256][257] cost, pred-major
    float* u        = C + (size_t)NN * NS;      // row (object) potentials
    float* sp_l     = u + NN;                   // sp shadow (scattered reads)
    int*   path     = (int*)(sp_l + NN);        // predecessor row per column
    int*   col4row  = path + NN;
    int*   row4col  = col4row + NN;
    int*   leftover = row4col + NN;

    const int  lane = threadIdx.x;              // 0..31
    const int  b    = blockIdx.x;
    const size_t base = (size_t)b * NN * NN;

    // ---- stage cost[b] into LDS (row stride 257 dwords) --------------------
    bool staged = false;
#if defined(__gfx1250__) && __has_builtin(__builtin_amdgcn_tensor_load_to_lds)
    {
        uint64_t ga      = (uint64_t)(uintptr_t)(costs + base);
        uint32_t lds_off = (uint32_t)(uintptr_t)C;   // low 32 bits = LDS addr
        // D# group0: count=1 | lds_addr | global_addr[56:0] | type=2
        u32x4 g0 = { 1u, lds_off, (uint32_t)ga,
                     (uint32_t)(((ga >> 32) & 0x01FFFFFFu) | (2u << 30)) };
        // D# group1: data_size=4B, pad_enable, pad_interval=256dw, pad=1dw,
        // tensor_dim0=tensor_dim1=256, tile_dim0=tile_dim1=256, stride0=256
        i32x8 g1 = { (int)((2u << 16) | (1u << 20) | (7u << 22)),
                     0x01000000,   // tensor_dim0[15:0] << 16
                     0x01000000,   // tensor_dim1[15:0] << 16
                     0x01000000,   // tile_dim0 << 16
                     0x00000100,   // tile_dim1=256, tile_dim2=0
                     256,          // tensor_dim0_stride
                     0, 0 };
        i32x4 gz = {0, 0, 0, 0};
#if defined(__clang_major__) && (__clang_major__ >= 23)
        i32x8 gz8 = {0, 0, 0, 0, 0, 0, 0, 0};
        __builtin_amdgcn_tensor_load_to_lds(g0, g1, gz, gz, gz8, 0);
#else
        __builtin_amdgcn_tensor_load_to_lds(g0, g1, gz, gz, 0);
#endif
#if __has_builtin(__builtin_amdgcn_s_wait_tensorcnt)
        __builtin_amdgcn_s_wait_tensorcnt((short)0);
#endif
        staged = true;
    }
#endif
    if (!staged) {
        for (int idx = lane; idx < NN * NN; idx += 32) {
            int r = idx >> 8, c = idx & 255;
            C[r * NS + c] = costs[base + idx];
        }
    }

    // ---- nvalid (prefix mask); detect bool8 vs int32 device layout ---------
    unsigned md0 = *(const unsigned*)mask_raw;   // mask[0][0..3] always true
    int cnt = 0;
    if (md0 == 0x01010101u) {                    // numpy bool8
        const unsigned char* m8 = (const unsigned char*)mask_raw + (size_t)b * NN;
        #pragma unroll
        for (int k = 0; k < K; ++k) cnt += m8[lane + 32 * k] ? 1 : 0;
    } else {                                     // int32 0/1
        const int* m32 = (const int*)mask_raw + (size_t)b * NN;
        #pragma unroll
        for (int k = 0; k < K; ++k) cnt += m32[lane + 32 * k] ? 1 : 0;
    }
    #pragma unroll
    for (int off = 16; off > 0; off >>= 1) cnt += __shfl_xor(cnt, off, 32);
    const int nv = cnt;                          // identical in all lanes

    // ---- init --------------------------------------------------------------
    float vj[K];                                 // v for owned columns
    float spj[K];                                // shortest-path dist (reg copy)
    #pragma unroll
    for (int k = 0; k < K; ++k) {
        int r = lane + 32 * k;
        vj[k] = 0.0f;
        u[r] = 0.0f;
        col4row[r] = -1;
        row4col[r] = -1;
    }

    // ---- outer loop: one shortest augmenting path per row ------------------
    for (int cur = 0; cur < NN; ++cur) {
        unsigned SCm = 0, SRm = 0;               // per-lane bit k = col/row L+32k
        #pragma unroll
        for (int k = 0; k < K; ++k) spj[k] = INF_F;

        int   i      = cur;                      // uniform across lanes
        float minVal = 0.0f;
        int   sink   = -1;

        do {
            if ((i & 31) == lane) SRm |= 1u << (i >> 5);
            const float ui    = u[i];            // uniform LDS broadcast read
            const bool  valid = (i < nv);

            // relax owned columns: cmat[i][j] = valid ? cost[j][i] : 0
            #pragma unroll
            for (int k = 0; k < K; ++k) {
                int   j = lane + 32 * k;
                float c = valid ? C[(size_t)j * NS + i] : 0.0f;
                float r = ((minVal + c) - ui) - vj[k];   // exact ref. order
                if (!((SCm >> k) & 1u) && r < spj[k]) {
                    spj[k] = r;
                    sp_l[j] = r;                 // write-through shadow
                    path[j] = i;
                }
            }

            // local argmin over owned columns (k ascending -> lowest j on tie)
            float m   = INF_F;
            int   idx = lane;
            #pragma unroll
            for (int k = 0; k < K; ++k) {
                float mk = ((SCm >> k) & 1u) ? INF_F : spj[k];
                if (mk < m) { m = mk; idx = lane + 32 * k; }
            }
            // wave32 butterfly argmin, lowest-index tie-break (== jnp.argmin);
            // xor reduction leaves the global result in every lane.
            #pragma unroll
            for (int off = 16; off > 0; off >>= 1) {
                float ov = __shfl_xor(m,   off, 32);
                int   oi = __shfl_xor(idx, off, 32);
                if (ov < m || (ov == m && oi < idx)) { m = ov; idx = oi; }
            }

            if ((idx & 31) == lane) SCm |= 1u << (idx >> 5);
            int nxt = row4col[idx];              // uniform LDS read
            minVal = m;
            if (nxt < 0) sink = idx; else i = nxt;
        } while (sink < 0);

        // ---- dual potential updates (before augmentation, as in reference)
        #pragma unroll
        for (int k = 0; k < K; ++k) {
            int r = lane + 32 * k;
            if ((SRm >> k) & 1u) {
                float du;
                if (r == cur) du = minVal;
                else {
                    int cc = col4row[r];
                    cc = cc < 0 ? 0 : (cc > NN - 1 ? NN - 1 : cc);
                    du = minVal - sp_l[cc];
                }
                u[r] += du;
            }
            if ((SCm >> k) & 1u) vj[k] -= (minVal - spj[k]);
        }

        // ---- augment alternating path (lane 0; DS ops are wave-ordered)
        if (lane == 0) {
            int j = sink;
            for (;;) {
                int ii = path[j];
                row4col[j] = ii;
                int jn = col4row[ii];
                col4row[ii] = j;
                if (ii == cur) break;
                j = jn;
            }
        }
    }

    // ---- output: valid rows -> match; invalid rows -> unused cols ascending
    // reuse path[] as 'used' flags (per-wave DS ordering keeps this safe)
    #pragma unroll
    for (int k = 0; k < K; ++k) path[lane + 32 * k] = 0;
    #pragma unroll
    for (int k = 0; k < K; ++k) {
        int r = lane + 32 * k;
        if (r < nv) path[col4row[r]] = 1;        // permutation: no collisions
    }
    if (lane == 0) {
        int p = 0;
        for (int c = 0; c < NN; ++c) if (!path[c]) leftover[p++] = c;
    }
    #pragma unroll
    for (int k = 0; k < K; ++k) {
        int r = lane + 32 * k;                   // coalesced 32x4B per k
        out[(size_t)b * NN + r] = (r < nv) ? col4row[r] : leftover[r - nv];
    }
}

extern "C" void kernel_launch(void* const* d_in, const int* in_sizes, int n_in,
                              void* d_out, int out_size, void* d_ws, size_t ws_size,
                              hipStream_t stream) {
    (void)n_in; (void)d_ws; (void)ws_size; (void)out_size;
    const float* costs = (const float*)d_in[0];
    const void*  mask  = d_in[1];
    const int s0 = in_sizes[0];   // B*N*N
    const int s1 = in_sizes[1];   // B*N
    const int N  = s0 / s1;       // == 256 for this reference
    const int B  = s1 / N;

    // LDS: cost tile [256][257] f32 + u, sp shadow, path, col4row, row4col,
    // leftover (6 * 256 dwords) = 263168 + 6144 bytes.
    size_t shbytes = (size_t)NN * NS * sizeof(float) + (size_t)6 * NN * 4;

    Matcher_75419625718226_kernel<<<dim3(B), dim3(32), shbytes, stream>>>(
        costs, mask, (int*)d_out);
}